// TransOp_expm_78219944394989
// MI455X (gfx1250) — compile-verified
//
#include <hip/hip_runtime.h>

typedef float v2f __attribute__((ext_vector_type(2)));
typedef float v8f __attribute__((ext_vector_type(8)));

#define WMMA4(a, b, c) \
  __builtin_amdgcn_wmma_f32_16x16x4_f32(false, (a), false, (b), (short)0, (c), false, false)

constexpr int BATCH = 8192;
constexpr int DICT  = 64;
constexpr int NN    = 8;     // small matrix dim
constexpr int MC    = 16;    // mixing coefficients
constexpr int BT    = 8;     // batches per workgroup (one per wave)
constexpr int ST    = 2;     // dicts per wave (block-diagonal pair)
constexpr int STRIDE = 20;   // padded LDS row stride (dwords): conflict-free on 64 banks
constexpr float INV_SCALE = 1.0f / 16.0f;  // 2^-4: 2 matrix squarings + 2 vector doublings
constexpr int WKS = 704;     // per-wave scratch floats: 2 tiles (320) + 2 vectors (16) + pad

static __device__ __forceinline__ void lds_fence() {
  __asm__ volatile("s_wait_dscnt 0" ::: "memory");
}

__global__ __launch_bounds__(256) void transop_expm_kernel(
    const float* __restrict__ x, const float* __restrict__ c,
    const float* __restrict__ psi, float* __restrict__ out)
{
  __shared__ float psi_s[MC * ST * NN * NN];  // [m][d*64 + n*8 + k], 2048 f32
  __shared__ float c_s[BT * MC];              // 128 f32
  __shared__ float wk[BT * WKS];

  const int t    = threadIdx.x;
  const int wave = t >> 5;
  const int lane = t & 31;
  const int b    = blockIdx.x * BT + wave;
  const int s0   = blockIdx.y * ST;

  // ---- Stage psi slice with async global->LDS b128 (2 per thread) ----
  #pragma unroll
  for (int i = 0; i < 2; ++i) {
    int idx4 = t + 256 * i;          // 0..511 groups of 4 floats
    int m    = idx4 >> 5;            // 32 groups (128 floats) per m
    int rem  = (idx4 & 31) * 4;      // float offset in [d*64 + nk]
    int d    = rem >> 6;
    int nk   = rem & 63;
    unsigned goff = (unsigned)(((m * DICT + s0 + d) * 64 + nk) * 4);
    unsigned loff = (unsigned)(size_t)(&psi_s[idx4 * 4]);  // low 32 bits = LDS offset
    __asm__ volatile("global_load_async_to_lds_b128 %0, %1, %2"
                     :: "v"(loff), "v"(goff), "s"(psi) : "memory");
  }
  if (t < BT * MC)
    c_s[t] = c[(blockIdx.x * BT + (t >> 4)) * MC + (t & 15)];
  __asm__ volatile("s_wait_asynccnt 0" ::: "memory");
  __syncthreads();

  float* mat0 = &wk[wave * WKS];
  float* mat1 = mat0 + 320;
  float* xv   = mat0 + 640;
  float* zv   = mat0 + 656;

  // x segment for this wave's two dicts (16 floats)
  if (lane < 16) xv[lane] = x[b * (DICT * NN) + s0 * NN + lane];

  float cr[MC];
  #pragma unroll
  for (int m = 0; m < MC; ++m) cr[m] = c_s[wave * MC + m];

  // ---- Build G = blockdiag(A0, A1) * 2^-4 into mat0 ----
  #pragma unroll
  for (int r = 0; r < 8; ++r) {
    int e  = lane + 32 * r;
    int Mr = e >> 4, Nc = e & 15;
    float v = 0.0f;
    bool blk1 = (Mr >= 8) & (Nc >= 8);
    if (((Mr < 8) & (Nc < 8)) | blk1) {
      int d   = blk1 ? 1 : 0;
      int off = d * 64 + (Mr - d * 8) * 8 + (Nc - d * 8);
      #pragma unroll
      for (int m = 0; m < MC; ++m) v += cr[m] * psi_s[m * 128 + off];
      v *= INV_SCALE;
    }
    mat0[Mr * STRIDE + Nc] = v;
  }
  lds_fence();

  // Lane coordinate helpers
  const int Ma  = lane & 15;          // A-operand row
  const int h2  = (lane >> 4) << 1;   // K sub-offset from lane half
  const int Ncd = lane & 15;          // C/D column
  const int Mhi = (lane >> 4) * 8;    // C/D row base

  auto loadA = [&](const float* tile, v2f a[4]) {
    #pragma unroll
    for (int ch = 0; ch < 4; ++ch) {
      a[ch].x = tile[Ma * STRIDE + (4 * ch + h2 + 0)];
      a[ch].y = tile[Ma * STRIDE + (4 * ch + h2 + 1)];
    }
  };
  auto loadB = [&](const float* tile, v2f bb[4]) {
    #pragma unroll
    for (int ch = 0; ch < 4; ++ch) {
      bb[ch].x = tile[(4 * ch + h2 + 0) * STRIDE + Ncd];
      bb[ch].y = tile[(4 * ch + h2 + 1) * STRIDE + Ncd];
    }
  };
  auto loadCD = [&](const float* tile, float v[8]) {
    #pragma unroll
    for (int r = 0; r < 8; ++r) v[r] = tile[(r + Mhi) * STRIDE + Ncd];
  };
  auto storeCD = [&](float* tile, const float v[8]) {
    #pragma unroll
    for (int r = 0; r < 8; ++r) tile[(r + Mhi) * STRIDE + Ncd] = v[r];
  };
  auto matmul = [&](const v2f a[4], const v2f bb[4]) -> v8f {
    v8f acc = {};
    acc = WMMA4(a[0], bb[0], acc);
    acc = WMMA4(a[1], bb[1], acc);
    acc = WMMA4(a[2], bb[2], acc);
    acc = WMMA4(a[3], bb[3], acc);
    return acc;
  };

  // ---- Paterson–Stockmeyer degree-8 Taylor: 4 matmuls ----
  v2f ga[4], gb[4];
  float gcd[8];
  loadA(mat0, ga);
  loadB(mat0, gb);
  loadCD(mat0, gcd);

  v8f g2 = matmul(ga, gb);                 // G^2
  float g2cd[8];
  #pragma unroll
  for (int r = 0; r < 8; ++r) g2cd[r] = g2[r];
  storeCD(mat1, g2cd);
  lds_fence();

  v2f g2a[4], g2b[4];
  loadA(mat1, g2a);
  loadB(mat1, g2b);
  v8f g4 = matmul(g2a, g2b);               // G^4
  v8f g3 = matmul(ga, g2b);                // G^3

  // U = c4 I + c5 G + c6 G2 + c7 G3 + c8 G4
  float ucd[8], g3cd[8], g4cd[8];
  #pragma unroll
  for (int r = 0; r < 8; ++r) {
    g3cd[r] = g3[r];
    g4cd[r] = g4[r];
    float del = ((r + Mhi) == Ncd) ? 1.0f : 0.0f;
    ucd[r] = (1.0f / 24.0f) * del + (1.0f / 120.0f) * gcd[r] +
             (1.0f / 720.0f) * g2cd[r] + (1.0f / 5040.0f) * g3cd[r] +
             (1.0f / 40320.0f) * g4cd[r];
  }
  storeCD(mat0, g4cd);   // overwrite G (no longer needed in LDS)
  storeCD(mat1, ucd);    // overwrite G2
  lds_fence();

  v2f g4a[4], ub[4];
  loadA(mat0, g4a);
  loadB(mat1, ub);
  v8f w = matmul(g4a, ub);                 // G4 * U

  // P = I + G + G2/2 + G3/6 + W  ~= exp(G)
  float pcd[8];
  #pragma unroll
  for (int r = 0; r < 8; ++r) {
    float del = ((r + Mhi) == Ncd) ? 1.0f : 0.0f;
    pcd[r] = del + gcd[r] + 0.5f * g2cd[r] + (1.0f / 6.0f) * g3cd[r] + w[r];
  }

  // ---- 2 squarings: exp(G)^4 = exp(A/4) ----
  #pragma unroll
  for (int sq = 0; sq < 2; ++sq) {
    storeCD(mat0, pcd);
    lds_fence();
    v2f pa[4], pb[4];
    loadA(mat0, pa);
    loadB(mat0, pb);
    v8f acc = matmul(pa, pb);
    #pragma unroll
    for (int r = 0; r < 8; ++r) pcd[r] = acc[r];
  }

  // ---- Apply T = exp(A/4) four times to x: y = exp(A) x ----
  storeCD(mat0, pcd);
  lds_fence();
  float* va = xv;
  float* vb = zv;
  #pragma unroll
  for (int it = 0; it < 4; ++it) {
    if (lane < 16) {
      const int p   = lane >> 3;
      const int row = lane & 7;
      const int Mr  = p * 8 + row;
      float acc = 0.0f;
      #pragma unroll
      for (int j = 0; j < 8; ++j)
        acc += mat0[Mr * STRIDE + p * 8 + j] * va[p * 8 + j];
      vb[lane] = acc;
    }
    lds_fence();
    float* tmp = va; va = vb; vb = tmp;
  }
  if (lane < 16)
    out[b * (DICT * NN) + s0 * NN + lane] = va[lane];
}

extern "C" void kernel_launch(void* const* d_in, const int* in_sizes, int n_in,
                              void* d_out, int out_size, void* d_ws, size_t ws_size,
                              hipStream_t stream) {
  (void)in_sizes; (void)n_in; (void)d_ws; (void)ws_size; (void)out_size;
  const float* x   = (const float*)d_in[0];  // (B, DICT*N) f32
  const float* c   = (const float*)d_in[1];  // (B, M) f32
  const float* psi = (const float*)d_in[2];  // (M, DICT, N, N) f32
  float* out = (float*)d_out;                // (B, DICT*N) f32

  dim3 grid(BATCH / BT, DICT / ST, 1);       // (1024, 32)
  dim3 block(256, 1, 1);                     // 8 waves, full EXEC for WMMA
  transop_expm_kernel<<<grid, block, 0, stream>>>(x, c, psi, out);
}